// Conv2D_11012296147198
// MI455X (gfx1250) — compile-verified
//
#include <hip/hip_runtime.h>
#include <stdint.h>

typedef __attribute__((ext_vector_type(2))) float v2f;
typedef __attribute__((ext_vector_type(4))) float v4f;
typedef __attribute__((ext_vector_type(8))) float v8f;
typedef __attribute__((ext_vector_type(4))) int   v4i;

#define IH 4096
#define IW 4096
#define KH 7
#define KW 7
#define OH (IH - KH + 1) /* 4090 */
#define OW (IW - KW + 1) /* 4090 */
#define TILES_X ((OW + 15) / 16)   /* 256 column tiles */
#define BROWS ((OH + 127) / 128)   /* 32 block rows (128 output rows per block) */
#define NKC 6                      /* K chunks of 4: k = 0..23 (band uses 0..21) */

#define PATCH_ROWS 134             /* 128 output rows + 6 halo */
#define LDS_STRIDE 28              /* dwords: >=24 needed; 28 -> 16B rows + no LDS bank conflicts */
#define CHUNKS_PER_ROW (LDS_STRIDE / 4)              /* 7 x b128 per row */
#define TOTAL_CHUNKS (PATCH_ROWS * CHUNKS_PER_ROW)   /* 938 */

#define AS1 __attribute__((address_space(1)))
#define AS3 __attribute__((address_space(3)))

__launch_bounds__(256)
__global__ void conv7x7_wmma_f32(const float* __restrict__ X,
                                 const float* __restrict__ Wt,
                                 const float* __restrict__ Bias,
                                 float* __restrict__ Out)
{
    __shared__ float tile[PATCH_ROWS * LDS_STRIDE];  /* 15008 B */

    const int tid  = threadIdx.x;
    const int lane = tid & 31;
    const int wave = tid >> 5;

    const int br    = blockIdx.x >> 8;   /* block row   0..31  */
    const int tc    = blockIdx.x & 255;  /* column tile 0..255 */
    const int cbase = tc * 16;

    int rstart = br * 128;
    if (rstart > IH - PATCH_ROWS) rstart = IH - PATCH_ROWS;  /* keep patch in-bounds */
    const int shift = br * 128 - rstart;                      /* 0, or 6 on last block row */

    /* ---- Stage the 134 x 28 input patch into LDS (coalesced 112B rows).
       Async global->LDS (ASYNCcnt path) when the toolchain exposes it. */
#pragma unroll
    for (int i = 0; i < 4; ++i) {
        const int idx = tid + i * 256;
        if (idx < TOTAL_CHUNKS) {
            const int r  = idx / CHUNKS_PER_ROW;
            const int cc = idx - r * CHUNKS_PER_ROW;
            const int gr = rstart + r;          /* always < IH by rstart clamp */
            int gc = cbase + cc * 4;
            if (gc > IW - 4) gc = IW - 4;       /* pollutes only masked outputs */
            const float* gp = X + (long)gr * IW + gc;
            float*       lp = &tile[r * LDS_STRIDE + cc * 4];
#if __has_builtin(__builtin_amdgcn_global_load_async_to_lds_b128)
            /* AS1 and generic share address values for global memory; re-tag via integer. */
            __builtin_amdgcn_global_load_async_to_lds_b128(
                (AS1 v4i*)(uintptr_t)gp, (AS3 v4i*)lp, /*offset=*/0, /*cpol=*/0);
#else
            *(v4f*)lp = *(const v4f*)gp;
#endif
        }
    }

    /* ---- Toeplitz B fragments from the 7x7 weights (uniform, register-resident).
       B_kh(k, j) = w(kh, k - j) for 0 <= k-j <= 6, else 0.
       Lane layout (4x16 f32 B): b[0]=B(4kc+2hi, n), b[1]=B(4kc+2hi+1, n). */
    const int n  = lane & 15;
    const int hi = lane >> 4;
    v2f bfrag[KH][NKC];
#pragma unroll
    for (int kh = 0; kh < KH; ++kh) {
#pragma unroll
        for (int kc = 0; kc < NKC; ++kc) {
            const int k0 = kc * 4 + 2 * hi;
            const int d0 = k0 - n;
            const int d1 = k0 + 1 - n;
            v2f b;
            b[0] = (d0 >= 0 && d0 < KW) ? Wt[kh * KW + d0] : 0.0f;
            b[1] = (d1 >= 0 && d1 < KW) ? Wt[kh * KW + d1] : 0.0f;
            bfrag[kh][kc] = b;
        }
    }

#if __has_builtin(__builtin_amdgcn_global_load_async_to_lds_b128)
#if __has_builtin(__builtin_amdgcn_s_wait_asynccnt)
    __builtin_amdgcn_s_wait_asynccnt(0);
#else
    asm volatile("s_wait_asynccnt 0x0" ::: "memory");
#endif
#endif
    __syncthreads();

    /* ---- Out_tile(16x16) = sum_kh  LDS_patch[w*16+kh .. +16, 0..24) x T_kh(24x16)
       A lane layout (16x4 f32 A): a[0]=A(n, 4kc+2hi), a[1]=A(n, 4kc+2hi+1).
       ds_load_b64 reads are bank-conflict-free at stride 28. */
    v8f acc = {0.0f, 0.0f, 0.0f, 0.0f, 0.0f, 0.0f, 0.0f, 0.0f};
#pragma unroll
    for (int kh = 0; kh < KH; ++kh) {
        int lr = wave * 16 + n + kh + shift;
        if (lr > PATCH_ROWS - 1) lr = PATCH_ROWS - 1;  /* pollutes only masked outputs */
        const float* lrow = &tile[lr * LDS_STRIDE];
#pragma unroll
        for (int kc = 0; kc < NKC; ++kc) {
            const v2f a = *(const v2f*)(lrow + kc * 4 + 2 * hi);
            acc = __builtin_amdgcn_wmma_f32_16x16x4_f32(
                /*neg_a=*/false, a,
                /*neg_b=*/false, bfrag[kh][kc],
                /*c_mod=*/(short)0, acc,
                /*reuse_a=*/false, /*reuse_b=*/false);
        }
    }

    /* ---- Store D: lane holds (row = v + 8*hi, col = n), v = 0..7 */
    const float bs    = Bias[0];
    const int   rbase = br * 128 + wave * 16;
    const int   ocol  = cbase + n;
    if (ocol < OW) {
#pragma unroll
        for (int v = 0; v < 8; ++v) {
            const int orow = rbase + v + 8 * hi;
            if (orow < OH) {
                Out[(long)orow * OW + ocol] = acc[v] + bs;
            }
        }
    }
}

extern "C" void kernel_launch(void* const* d_in, const int* in_sizes, int n_in,
                              void* d_out, int out_size, void* d_ws, size_t ws_size,
                              hipStream_t stream) {
    const float* X    = (const float*)d_in[0];  /* 4096*4096 f32 */
    const float* Wt   = (const float*)d_in[1];  /* 7*7 f32 */
    const float* Bias = (const float*)d_in[2];  /* 1 f32 */
    float*       Out  = (float*)d_out;          /* 4090*4090 f32 */

    const int blocks = BROWS * TILES_X;  /* 32 * 256 = 8192 */
    conv7x7_wmma_f32<<<blocks, 256, 0, stream>>>(X, Wt, Bias, Out);
}